// TSApproximation_27693949124951
// MI455X (gfx1250) — compile-verified
//
#include <hip/hip_runtime.h>
#include <hip/hip_bf16.h>

// out[i] = bias + sum_j c[j,0]*x[i,0]^j + c[j,1]*x[i,1]^j   (N = 4096*4096, T = 8)
// Memory-bound: 12 B/pixel, ~2.5 FLOP/B -> HBM roofline ~8.6 us on MI455X.
// Strategy: stream x via async global->LDS (ASYNCcnt, double buffer), Horner in
// VALU with SGPR-resident coefficients, non-temporal b128 stores.

typedef float v4f __attribute__((ext_vector_type(4)));
typedef int i4v __attribute__((vector_size(16)));  // matches builtin's V4i pointee

#define THREADS 256
#define PX_PER_THREAD 4
#define TILE_PX (THREADS * PX_PER_THREAD)   // 1024 pixels per tile
#define TILE_FLOATS (TILE_PX * 2)           // 2048 floats = 8 KB per tile
#define NTERMS 8

#if defined(__AMDGCN__) && \
    __has_builtin(__builtin_amdgcn_global_load_async_to_lds_b128) && \
    __has_builtin(__builtin_amdgcn_s_wait_asynccnt)
#define USE_ASYNC_LDS 1
typedef __attribute__((address_space(1))) i4v* gb128_p;  // global int4*
typedef __attribute__((address_space(3))) i4v* lb128_p;  // LDS int4*
#else
#define USE_ASYNC_LDS 0
#endif

__device__ __forceinline__ float horner8(const float (&c)[NTERMS], float v) {
  float p = c[NTERMS - 1];
#pragma unroll
  for (int j = NTERMS - 2; j >= 0; --j) p = __builtin_fmaf(p, v, c[j]);
  return p;
}

__global__ __launch_bounds__(THREADS) void poly_main(
    const float* __restrict__ x, const float* __restrict__ coef,
    const float* __restrict__ bias, float* __restrict__ out, int num_tiles) {
  // Uniform loads -> s_load into SGPRs (coefficients live in scalar regs).
  float c0[NTERMS], c1[NTERMS];
#pragma unroll
  for (int j = 0; j < NTERMS; ++j) {
    c0[j] = coef[2 * j];
    c1[j] = coef[2 * j + 1];
  }
  const float bb = bias[0];

  const int tid = threadIdx.x;
  const int stride = gridDim.x;

#if USE_ASYNC_LDS
  __shared__ __attribute__((aligned(16))) float lds[2][TILE_FLOATS];

  auto stage = [&](int buf, int tile) {
    const float* g = x + (size_t)tile * TILE_FLOATS + (size_t)tid * 8;
    float* l = &lds[buf][tid * 8];
    // INST_OFFSET applies to both global and LDS addresses (ISA 15.18.3).
    __builtin_amdgcn_global_load_async_to_lds_b128((gb128_p)g, (lb128_p)l, 0, 0);
    __builtin_amdgcn_global_load_async_to_lds_b128((gb128_p)g, (lb128_p)l, 16, 0);
  };

  int tile = blockIdx.x;
  if (tile < num_tiles) stage(0, tile);
  int cur = 0;
  for (; tile < num_tiles; tile += stride) {
    const int nxt = tile + stride;
    if (nxt < num_tiles) {
      stage(cur ^ 1, nxt);                    // prefetch next tile (2 more ops)
      __builtin_amdgcn_s_wait_asynccnt(2);    // retire current tile's 2 ops
    } else {
      __builtin_amdgcn_s_wait_asynccnt(0);
    }
    asm volatile("" ::: "memory");            // keep ds_loads below the wait

    const v4f a = *(const v4f*)&lds[cur][tid * 8];      // px0:(a.x,a.y) px1:(a.z,a.w)
    const v4f b = *(const v4f*)&lds[cur][tid * 8 + 4];  // px2:(b.x,b.y) px3:(b.z,b.w)
    v4f o;
    o.x = bb + horner8(c0, a.x) + horner8(c1, a.y);
    o.y = bb + horner8(c0, a.z) + horner8(c1, a.w);
    o.z = bb + horner8(c0, b.x) + horner8(c1, b.y);
    o.w = bb + horner8(c0, b.z) + horner8(c1, b.w);
    __builtin_nontemporal_store(o, (v4f*)(out + (size_t)tile * TILE_PX) + tid);
    cur ^= 1;
  }
#else
  // Fallback: direct non-temporal b128 loads (still hits the HBM roofline).
  for (int tile = blockIdx.x; tile < num_tiles; tile += stride) {
    const v4f* xp = (const v4f*)(x + (size_t)tile * TILE_FLOATS) + tid * 2;
    const v4f a = __builtin_nontemporal_load(xp);
    const v4f b = __builtin_nontemporal_load(xp + 1);
    v4f o;
    o.x = bb + horner8(c0, a.x) + horner8(c1, a.y);
    o.y = bb + horner8(c0, a.z) + horner8(c1, a.w);
    o.z = bb + horner8(c0, b.x) + horner8(c1, b.y);
    o.w = bb + horner8(c0, b.z) + horner8(c1, b.w);
    __builtin_nontemporal_store(o, (v4f*)(out + (size_t)tile * TILE_PX) + tid);
  }
#endif
}

// Tail for any pixels not covered by whole tiles (N = 16,777,216 divides evenly,
// but keep it robust).
__global__ void poly_tail(const float* __restrict__ x, const float* __restrict__ coef,
                          const float* __restrict__ bias, float* __restrict__ out,
                          int start, int n) {
  int i = start + blockIdx.x * blockDim.x + threadIdx.x;
  if (i >= n) return;
  float c0[NTERMS], c1[NTERMS];
#pragma unroll
  for (int j = 0; j < NTERMS; ++j) {
    c0[j] = coef[2 * j];
    c1[j] = coef[2 * j + 1];
  }
  out[i] = bias[0] + horner8(c0, x[2 * i]) + horner8(c1, x[2 * i + 1]);
}

extern "C" void kernel_launch(void* const* d_in, const int* in_sizes, int n_in,
                              void* d_out, int out_size, void* d_ws, size_t ws_size,
                              hipStream_t stream) {
  const float* x = (const float*)d_in[0];     // (N, 2) fp32
  const float* coef = (const float*)d_in[1];  // (8, 2) fp32
  const float* bias = (const float*)d_in[2];  // (1,)  fp32
  float* out = (float*)d_out;                 // N fp32

  const int n = out_size;                     // number of pixels
  const int num_tiles = n / TILE_PX;
  const int start = num_tiles * TILE_PX;

  if (num_tiles > 0) {
    int blocks = num_tiles < 4096 ? num_tiles : 4096;  // ~4 tiles/block: 2-deep pipeline amortized
    poly_main<<<blocks, THREADS, 0, stream>>>(x, coef, bias, out, num_tiles);
  }
  const int rem = n - start;
  if (rem > 0) {
    poly_tail<<<(rem + THREADS - 1) / THREADS, THREADS, 0, stream>>>(x, coef, bias, out, start, n);
  }
}